// WHTConv2D_24318104830353
// MI455X (gfx1250) — compile-verified
//
#include <hip/hip_runtime.h>

typedef __attribute__((ext_vector_type(2))) float v2f;
typedef __attribute__((ext_vector_type(8))) float v8f;

#define Hcrop 56
#define Wcrop 56
#define HPAD  64
#define WPAD  64
#define BATCH 64
#define CH    128
#define NTOT  (BATCH * HPAD * WPAD)   // 262144 GEMM columns
#define NTILE 64                      // columns per block in GEMM kernel
#define LDSW  (WPAD + 1)              // padded row stride (bank-conflict-free cols)

// ---------------------------------------------------------------------------
// In-LDS 64x64 Walsh-Hadamard transform helpers (256-thread block).
// ---------------------------------------------------------------------------
__device__ __forceinline__ void fwht_rows(float* t, int tid) {
  for (int s = 1; s < WPAD; s <<= 1) {
    for (int idx = tid; idx < HPAD * (WPAD / 2); idx += 256) {
      int row = idx >> 5;
      int pj  = idx & 31;
      int i   = ((pj & ~(s - 1)) << 1) | (pj & (s - 1));
      float* base = &t[row * LDSW];
      float a = base[i];
      float b = base[i + s];
      base[i]     = a + b;
      base[i + s] = a - b;
    }
    __syncthreads();
  }
}

__device__ __forceinline__ void fwht_cols(float* t, int tid) {
  for (int s = 1; s < HPAD; s <<= 1) {
    for (int idx = tid; idx < WPAD * (HPAD / 2); idx += 256) {
      int col = idx >> 5;
      int pj  = idx & 31;
      int i   = ((pj & ~(s - 1)) << 1) | (pj & (s - 1));
      float a = t[i * LDSW + col];
      float b = t[(i + s) * LDSW + col];
      t[i * LDSW + col]       = a + b;
      t[(i + s) * LDSW + col] = a - b;
    }
    __syncthreads();
  }
}

// ---------------------------------------------------------------------------
// Kernel 1: pad + forward 2-D FWHT, store channel-major spectral buffer
//   f2[c * NTOT + b * 4096 + h*64 + w]
// ---------------------------------------------------------------------------
__global__ void __launch_bounds__(256) fwht_fwd_kernel(const float* __restrict__ x,
                                                       float* __restrict__ f2) {
  __shared__ float t[HPAD * LDSW];
  const int tid = threadIdx.x;
  const int img = blockIdx.x;          // b*CH + c
  const int b   = img / CH;
  const int c   = img % CH;

  const float* xi = x + (size_t)img * (Hcrop * Wcrop);
  for (int i = tid; i < HPAD * WPAD; i += 256) {
    int h = i >> 6, w = i & 63;
    float val = (h < Hcrop && w < Wcrop) ? xi[h * Wcrop + w] : 0.0f;
    t[h * LDSW + w] = val;
  }
  __syncthreads();

  fwht_rows(t, tid);
  fwht_cols(t, tid);

  float* out = f2 + (size_t)c * NTOT + (size_t)b * (HPAD * WPAD);
  for (int i = tid; i < HPAD * WPAD; i += 256) {
    int h = i >> 6, w = i & 63;
    out[i] = t[h * LDSW + w];
  }
}

// ---------------------------------------------------------------------------
// Kernel 2: per-column spectral channel mix via V_WMMA_F32_16X16X4_F32,
// branch scaling by v[p], soft-threshold with relu(T[p]), branch sum.
// Operates IN PLACE on the spectral buffer (panel staged to LDS first).
// ---------------------------------------------------------------------------
__global__ void __launch_bounds__(256, 1) spectral_gemm_kernel(
    float* __restrict__ f2,            // [CH][NTOT], read then overwritten
    const float* __restrict__ convw,   // [2][128][128]  (o, i) row-major
    const float* __restrict__ vmask,   // [2][64][64]
    const float* __restrict__ Tthr) {  // [2][64][64]
  __shared__ float sf2[CH * NTILE];    // 32 KB B-panel: [k][col]

  const int tid = threadIdx.x;
  const int n0  = blockIdx.x * NTILE;  // this block's first column

  // Stage the full K x NTILE panel (coalesced 256B rows).
  for (int i = tid; i < CH * NTILE; i += 256) {
    int k = i >> 6, col = i & 63;
    sf2[i] = f2[(size_t)k * NTOT + (n0 + col)];
  }
  __syncthreads();

  const int lane = tid & 31;
  const int wave = tid >> 5;           // M-tile 0..7  -> rows 16*wave..+15
  const int m0   = wave * 16;
  const int lrow = lane & 15;          // M index within A-frag / N within B-frag
  const int lhi  = lane >> 4;          // selects K pair {0,1} vs {2,3}

  const float* w0 = convw + (size_t)(m0 + lrow) * CH;             // branch 0
  const float* w1 = convw + (size_t)(CH * CH) + (size_t)(m0 + lrow) * CH;

  v8f acc[2][4];
  v8f zero = {};
#pragma unroll
  for (int p = 0; p < 2; ++p)
#pragma unroll
    for (int ch = 0; ch < 4; ++ch) acc[p][ch] = zero;

  for (int k0 = 0; k0 < CH; k0 += 4) {
    const int ka = k0 + 2 * lhi;       // lanes 0-15: K=k0,k0+1; lanes 16-31: k0+2,k0+3
    __builtin_prefetch(&w0[ka + 4], 0, 3);
    __builtin_prefetch(&w1[ka + 4], 0, 3);
    v2f A0, A1;
    A0.x = w0[ka];
    A0.y = w0[ka + 1];
    A1.x = w1[ka];
    A1.y = w1[ka + 1];
#pragma unroll
    for (int ch = 0; ch < 4; ++ch) {
      const int col = ch * 16 + lrow;
      v2f B;
      B.x = sf2[ka * NTILE + col];
      B.y = sf2[(ka + 1) * NTILE + col];
      acc[0][ch] = __builtin_amdgcn_wmma_f32_16x16x4_f32(
          false, A0, false, B, (short)0, acc[0][ch], false, false);
      acc[1][ch] = __builtin_amdgcn_wmma_f32_16x16x4_f32(
          false, A1, false, B, (short)0, acc[1][ch], false, false);
    }
  }

  // Epilogue: columns n0..n0+63 all share one h; w == column index.
  const int h = (n0 >> 6) & 63;
#pragma unroll
  for (int ch = 0; ch < 4; ++ch) {
    const int colw = ch * 16 + lrow;
    const int hw   = h * 64 + colw;
    const float v0 = vmask[hw];
    const float v1 = vmask[4096 + hw];
    const float t0 = fmaxf(Tthr[hw], 0.0f);
    const float t1 = fmaxf(Tthr[4096 + hw], 0.0f);
#pragma unroll
    for (int r = 0; r < 8; ++r) {
      const int m = m0 + r + 8 * lhi;  // C/D layout: vgpr r, lane>=16 -> M+8
      float x0 = v0 * acc[0][ch][r];
      float a0 = fabsf(x0) - t0;
      float s0 = (a0 > 0.0f) ? copysignf(a0, x0) : 0.0f;
      float x1 = v1 * acc[1][ch][r];
      float a1 = fabsf(x1) - t1;
      float s1 = (a1 > 0.0f) ? copysignf(a1, x1) : 0.0f;
      f2[(size_t)m * NTOT + (n0 + colw)] = s0 + s1;
    }
  }
}

// ---------------------------------------------------------------------------
// Kernel 3: inverse 2-D FWHT (scale 1/4096), crop, residual add.
// ---------------------------------------------------------------------------
__global__ void __launch_bounds__(256) fwht_inv_kernel(const float* __restrict__ f6,
                                                       const float* __restrict__ x,
                                                       float* __restrict__ y) {
  __shared__ float t[HPAD * LDSW];
  const int tid = threadIdx.x;
  const int img = blockIdx.x;          // b*CH + o
  const int b   = img / CH;
  const int o   = img % CH;

  const float* in = f6 + (size_t)o * NTOT + (size_t)b * (HPAD * WPAD);
  for (int i = tid; i < HPAD * WPAD; i += 256) {
    t[(i >> 6) * LDSW + (i & 63)] = in[i];
  }
  __syncthreads();

  fwht_rows(t, tid);
  fwht_cols(t, tid);

  const float* xi = x + (size_t)img * (Hcrop * Wcrop);
  float* yo       = y + (size_t)img * (Hcrop * Wcrop);
  const float scale = 1.0f / 4096.0f;
  for (int i = tid; i < Hcrop * Wcrop; i += 256) {
    int h = i / Wcrop, w = i % Wcrop;
    yo[i] = t[h * LDSW + w] * scale + xi[i];
  }
}

// ---------------------------------------------------------------------------
extern "C" void kernel_launch(void* const* d_in, const int* in_sizes, int n_in,
                              void* d_out, int out_size, void* d_ws, size_t ws_size,
                              hipStream_t stream) {
  const float* x     = (const float*)d_in[0];
  const float* convw = (const float*)d_in[1];
  const float* vmask = (const float*)d_in[2];
  const float* Tthr  = (const float*)d_in[3];
  float* out = (float*)d_out;

  float* f2 = (float*)d_ws;  // CH * NTOT floats = 128 MiB spectral buffer

  fwht_fwd_kernel<<<BATCH * CH, 256, 0, stream>>>(x, f2);
  spectral_gemm_kernel<<<NTOT / NTILE, 256, 0, stream>>>(f2, convw, vmask, Tthr);
  fwht_inv_kernel<<<BATCH * CH, 256, 0, stream>>>(f2, x, out);
}